// GraphAttentionLayer_4114578670033
// MI455X (gfx1250) — compile-verified
//
#include <hip/hip_runtime.h>

// GAT layer: N=8192, IN_F=256, OUT_F=128
// d_in: x (N*IN_F f32), adj (N*N i32), W (IN_F*OUT_F f32), a (2*OUT_F f32)
// d_out: [ out (N*OUT_F f32) | attention (N*N f32) ]
// d_ws : h (N*OUT_F) | f1 (N) | f2 (N) | denom (N)   (~4.1 MB)

#define N_NODES 8192
#define IN_FEAT 256
#define OUT_FEAT 128
#define NEG_SLOPE 0.01f
#define ROWS_PER_BLK 64
#define JCHUNK 128

typedef __attribute__((ext_vector_type(2))) float v2f;
typedef __attribute__((ext_vector_type(8))) float v8f;

__device__ __forceinline__ v8f wmma_f32(v2f a, v2f b, v8f c) {
  // D = A(16x4 f32) * B(4x16 f32) + C(16x16 f32)
  return __builtin_amdgcn_wmma_f32_16x16x4_f32(
      /*neg_a=*/false, a, /*neg_b=*/false, b,
      /*c_mod=*/(short)0, c, /*reuse_a=*/false, /*reuse_b=*/false);
}

// ---------------- Kernel 1: h = x @ W (f32 WMMA) ----------------
// grid = N/16 blocks, 256 threads (8 waves). Wave w owns N-tile cols [16w,16w+16).
__global__ __launch_bounds__(256) void gemm_xw(const float* __restrict__ x,
                                               const float* __restrict__ W,
                                               float* __restrict__ h) {
  const int lane = threadIdx.x & 31;
  const int wave = threadIdx.x >> 5;
  const int row0 = blockIdx.x * 16;
  const int m16  = lane & 15;          // A row / B,D col within tile
  const int half = lane >> 4;          // selects K pair per ISA layout
  const int ncol = wave * 16 + m16;

  v8f c = {};
  const float* xrow = x + (size_t)(row0 + m16) * IN_FEAT;
  for (int k = 0; k < IN_FEAT; k += 4) {
    v2f a = *(const v2f*)(xrow + k + 2 * half);       // A[m][k+2h], A[m][k+2h+1]
    const int kr = k + 2 * half;
    v2f b;
    b.x = W[(size_t)kr * OUT_FEAT + ncol];            // B[k+2h][n]
    b.y = W[(size_t)(kr + 1) * OUT_FEAT + ncol];      // B[k+2h+1][n]
    c = wmma_f32(a, b, c);
  }
#pragma unroll
  for (int r = 0; r < 8; ++r)
    h[(size_t)(row0 + r + 8 * half) * OUT_FEAT + ncol] = c[r];
}

// ---------------- Kernel 2: f1 = h@a1, f2 = h@a2 ----------------
__global__ __launch_bounds__(256) void fvec(const float* __restrict__ h,
                                            const float* __restrict__ a,
                                            float* __restrict__ f1,
                                            float* __restrict__ f2) {
  const int i = blockIdx.x * blockDim.x + threadIdx.x;
  if (i >= N_NODES) return;
  const float* hr = h + (size_t)i * OUT_FEAT;
  float s1 = 0.f, s2 = 0.f;
  for (int c = 0; c < OUT_FEAT; ++c) {
    const float v = hr[c];
    s1 += v * a[c];
    s2 += v * a[OUT_FEAT + c];
  }
  f1[i] = s1;
  f2[i] = s2;
}

// ---------------- Kernel 3: per-row softmax denominator ----------------
// One block per row; single pass over adj row.
__global__ __launch_bounds__(256) void row_denom(const int* __restrict__ adj,
                                                 const float* __restrict__ f1,
                                                 const float* __restrict__ f2,
                                                 float* __restrict__ denom) {
  __shared__ float red[256];
  const int i = blockIdx.x;
  const float fi = f1[i];
  const size_t base = (size_t)i * N_NODES;
  float s = 0.f;
  for (int j = threadIdx.x; j < N_NODES; j += 256) {
    if (adj[base + j] > 0) {
      float e = fi + f2[j];
      e = e > 0.f ? e : e * NEG_SLOPE;
      s += __expf(e);
    }
  }
  red[threadIdx.x] = s;
  __syncthreads();
  for (int off = 128; off > 0; off >>= 1) {
    if (threadIdx.x < off) red[threadIdx.x] += red[threadIdx.x + off];
    __syncthreads();
  }
  if (threadIdx.x == 0) denom[i] = red[0];
}

// ---------------- Kernel 4: attention write + out = attention @ h ----------------
// Each block: 64 rows. LDS 64x128 normalized-attention tile -> A operand.
// 8 waves x 4 M-subtiles of f32 WMMA accumulators; B fragment reused 4x.
__global__ __launch_bounds__(256) void attn_out(const int* __restrict__ adj,
                                                const float* __restrict__ h,
                                                const float* __restrict__ f1,
                                                const float* __restrict__ f2,
                                                const float* __restrict__ denom,
                                                float* __restrict__ out,
                                                float* __restrict__ att) {
  __shared__ float att_tile[ROWS_PER_BLK * JCHUNK];  // 32 KB
  __shared__ float s_f1[ROWS_PER_BLK];
  __shared__ float s_scale[ROWS_PER_BLK];            // 1/denom, or -1 => uniform row

  const int t    = threadIdx.x;
  const int lane = t & 31;
  const int wave = t >> 5;
  const int row0 = blockIdx.x * ROWS_PER_BLK;

  if (t < ROWS_PER_BLK) {
    s_f1[t] = f1[row0 + t];
    const float d = denom[row0 + t];
    s_scale[t] = d > 0.f ? (1.f / d) : -1.f;
  }
  __syncthreads();

  const int m16  = lane & 15;
  const int half = lane >> 4;
  const int ncol = wave * 16 + m16;

  v8f c0 = {}, c1 = {}, c2 = {}, c3 = {};

  for (int jc = 0; jc < N_NODES; jc += JCHUNK) {
    // Build normalized attention tile (and emit it to the output matrix).
#pragma unroll
    for (int r = 0; r < (ROWS_PER_BLK * JCHUNK) / 256; ++r) {
      const int idx  = t + 256 * r;
      const int m    = idx >> 7;       // / JCHUNK
      const int colj = idx & (JCHUNK - 1);
      const int j    = jc + colj;
      const float sc = s_scale[m];
      float v;
      if (sc < 0.f) {                  // fully-masked row -> uniform softmax
        v = 1.0f / (float)N_NODES;
      } else if (adj[(size_t)(row0 + m) * N_NODES + j] > 0) {
        float e = s_f1[m] + f2[j];
        e = e > 0.f ? e : e * NEG_SLOPE;
        v = __expf(e) * sc;
      } else {
        v = 0.f;
      }
      att_tile[idx] = v;
      att[(size_t)(row0 + m) * N_NODES + j] = v;
    }
    __syncthreads();  // EXEC fully reconverged before WMMA

    // out += att_tile(64 x 128) @ h(jc:jc+128, :), wave handles 16 cols.
    for (int kk = 0; kk < JCHUNK; kk += 4) {
      const int kr = jc + kk + 2 * half;
      v2f b;
      b.x = h[(size_t)kr * OUT_FEAT + ncol];
      b.y = h[(size_t)(kr + 1) * OUT_FEAT + ncol];
      v2f a;
      a  = *(const v2f*)(att_tile + (0 * 16 + m16) * JCHUNK + kk + 2 * half);
      c0 = wmma_f32(a, b, c0);
      a  = *(const v2f*)(att_tile + (1 * 16 + m16) * JCHUNK + kk + 2 * half);
      c1 = wmma_f32(a, b, c1);
      a  = *(const v2f*)(att_tile + (2 * 16 + m16) * JCHUNK + kk + 2 * half);
      c2 = wmma_f32(a, b, c2);
      a  = *(const v2f*)(att_tile + (3 * 16 + m16) * JCHUNK + kk + 2 * half);
      c3 = wmma_f32(a, b, c3);
    }
    __syncthreads();  // protect att_tile before next chunk overwrites it
  }

#pragma unroll
  for (int r = 0; r < 8; ++r) {
    const int rr = r + 8 * half;
    out[(size_t)(row0 +  0 + rr) * OUT_FEAT + ncol] = c0[r];
    out[(size_t)(row0 + 16 + rr) * OUT_FEAT + ncol] = c1[r];
    out[(size_t)(row0 + 32 + rr) * OUT_FEAT + ncol] = c2[r];
    out[(size_t)(row0 + 48 + rr) * OUT_FEAT + ncol] = c3[r];
  }
}

extern "C" void kernel_launch(void* const* d_in, const int* in_sizes, int n_in,
                              void* d_out, int out_size, void* d_ws, size_t ws_size,
                              hipStream_t stream) {
  (void)in_sizes; (void)n_in; (void)out_size; (void)ws_size;
  const float* x   = (const float*)d_in[0];
  const int*   adj = (const int*)d_in[1];
  const float* W   = (const float*)d_in[2];
  const float* a   = (const float*)d_in[3];

  float* out = (float*)d_out;
  float* att = out + (size_t)N_NODES * OUT_FEAT;

  float* h     = (float*)d_ws;
  float* f1    = h + (size_t)N_NODES * OUT_FEAT;
  float* f2    = f1 + N_NODES;
  float* denom = f2 + N_NODES;

  gemm_xw<<<N_NODES / 16, 256, 0, stream>>>(x, W, h);
  fvec<<<(N_NODES + 255) / 256, 256, 0, stream>>>(h, a, f1, f2);
  row_denom<<<N_NODES, 256, 0, stream>>>(adj, f1, f2, denom);
  attn_out<<<N_NODES / ROWS_PER_BLK, 256, 0, stream>>>(adj, h, f1, f2, denom, out, att);
}